// LotkaVolterra_23012434772350
// MI455X (gfx1250) — compile-verified
//
#include <hip/hip_runtime.h>

// Lotka-Volterra RK4 trajectory kernel for MI455X (gfx1250).
//
// Roofline: output = 2*2001*65536 fp32 = 1.05 GB written once -> ~45us floor
// at 23.3 TB/s HBM. Math is ~34 packed-fp32 ops per step per float2 column
// pair -> ~27us of VALU on ~1k SIMD32s, so the kernel is store-bound.
// Strategy:
//   * 2 batch columns per thread, state held as float2 in VGPR pairs.
//   * Core math via explicit CDNA5 VOP3P packed-fp32 inline asm
//     (v_pk_fma_f32 / v_pk_mul_f32 / v_pk_add_f32) -> 2x fp32 rate.
//   * Non-temporal global_store_b64 for the trajectory (1.05 GB >> 192 MB L2,
//     never re-read) so we don't thrash L2.
//   * 32768 threads = 1024 wave32 -> covers all WGPs.

typedef __attribute__((ext_vector_type(2))) float f32x2;

#define LV_NSTEPS 2000
#define LV_BATCH  65536
#define LV_TOUT   (LV_NSTEPS + 1)

static __device__ __forceinline__ f32x2 pk_mul(f32x2 a, f32x2 b) {
    f32x2 d;
    asm("v_pk_mul_f32 %0, %1, %2" : "=v"(d) : "v"(a), "v"(b));
    return d;
}
static __device__ __forceinline__ f32x2 pk_add(f32x2 a, f32x2 b) {
    f32x2 d;
    asm("v_pk_add_f32 %0, %1, %2" : "=v"(d) : "v"(a), "v"(b));
    return d;
}
static __device__ __forceinline__ f32x2 pk_fma(f32x2 a, f32x2 b, f32x2 c) {
    f32x2 d;
    asm("v_pk_fma_f32 %0, %1, %2, %3" : "=v"(d) : "v"(a), "v"(b), "v"(c));
    return d;
}

__global__ __launch_bounds__(128) void lv_rk4_kernel(
    const float* __restrict__ x0,
    const float* __restrict__ alpha,
    const float* __restrict__ beta,
    const float* __restrict__ gma,
    const float* __restrict__ dlt,
    float* __restrict__ out)
{
    const int tid = blockIdx.x * blockDim.x + threadIdx.x;
    const int b = tid * 2;   // grid exactly covers BATCH/2 threads

    const float a  = alpha[0];
    const float bb = beta[0];
    const float g  = gma[0];
    const float d  = dlt[0];

    const float dt  = 40.0f / (float)LV_NSTEPS;   // 0.02
    const float hdt = 0.5f * dt;
    const float dt6 = dt * (1.0f / 6.0f);

    // packed constants (VGPR pairs, loop-invariant)
    const f32x2 av   = {a, a};
    const f32x2 nbv  = {-bb, -bb};
    const f32x2 dv   = {d, d};
    const f32x2 ngv  = {-g, -g};
    const f32x2 hdtv = {hdt, hdt};
    const f32x2 dtv  = {dt, dt};
    const f32x2 dt6v = {dt6, dt6};
    const f32x2 twov = {2.0f, 2.0f};

    // state: 2 batch columns packed per thread
    f32x2 prey = *(const f32x2*)(x0 + b);
    f32x2 pred = *(const f32x2*)(x0 + LV_BATCH + b);

    // out[(0*TOUT + t)*BATCH + b] = prey, out[(1*TOUT + t)*BATCH + b] = pred
    float* oprey = out + b;
    float* opred = out + (size_t)LV_TOUT * LV_BATCH + b;

    // t = 0: initial condition
    __builtin_nontemporal_store(prey, (f32x2*)oprey);
    __builtin_nontemporal_store(pred, (f32x2*)opred);
    oprey += LV_BATCH;
    opred += LV_BATCH;

    for (int t = 0; t < LV_NSTEPS; ++t) {
        // k1 = f(x):  kp = a*prey - b*prey*pred,  kq = d*prey*pred - g*pred
        f32x2 pp  = pk_mul(prey, pred);
        f32x2 k1p = pk_fma(av, prey, pk_mul(nbv, pp));
        f32x2 k1q = pk_fma(dv, pp,  pk_mul(ngv, pred));

        // k2 = f(x + hdt*k1)
        f32x2 p2 = pk_fma(hdtv, k1p, prey);
        f32x2 q2 = pk_fma(hdtv, k1q, pred);
        pp = pk_mul(p2, q2);
        f32x2 k2p = pk_fma(av, p2, pk_mul(nbv, pp));
        f32x2 k2q = pk_fma(dv, pp, pk_mul(ngv, q2));

        // k3 = f(x + hdt*k2)
        f32x2 p3 = pk_fma(hdtv, k2p, prey);
        f32x2 q3 = pk_fma(hdtv, k2q, pred);
        pp = pk_mul(p3, q3);
        f32x2 k3p = pk_fma(av, p3, pk_mul(nbv, pp));
        f32x2 k3q = pk_fma(dv, pp, pk_mul(ngv, q3));

        // k4 = f(x + dt*k3)
        f32x2 p4 = pk_fma(dtv, k3p, prey);
        f32x2 q4 = pk_fma(dtv, k3q, pred);
        pp = pk_mul(p4, q4);
        f32x2 k4p = pk_fma(av, p4, pk_mul(nbv, pp));
        f32x2 k4q = pk_fma(dv, pp, pk_mul(ngv, q4));

        // x += dt/6 * (k1 + 2*(k2+k3) + k4)
        f32x2 sp = pk_fma(twov, pk_add(k2p, k3p), pk_add(k1p, k4p));
        f32x2 sq = pk_fma(twov, pk_add(k2q, k3q), pk_add(k1q, k4q));
        prey = pk_fma(dt6v, sp, prey);
        pred = pk_fma(dt6v, sq, pred);

        __builtin_nontemporal_store(prey, (f32x2*)oprey);
        __builtin_nontemporal_store(pred, (f32x2*)opred);
        oprey += LV_BATCH;
        opred += LV_BATCH;
    }
}

extern "C" void kernel_launch(void* const* d_in, const int* in_sizes, int n_in,
                              void* d_out, int out_size, void* d_ws, size_t ws_size,
                              hipStream_t stream) {
    const float* x0    = (const float*)d_in[0];
    const float* alpha = (const float*)d_in[1];
    const float* beta  = (const float*)d_in[2];
    const float* gma   = (const float*)d_in[3];
    const float* dlt   = (const float*)d_in[4];
    float* out = (float*)d_out;

    const int threads = LV_BATCH / 2;    // 2 batch columns per thread
    const int block   = 128;             // 4 wave32 -> one WGP's 4 SIMD32s
    const int grid    = threads / block; // 256 workgroups, exact coverage

    lv_rk4_kernel<<<grid, block, 0, stream>>>(x0, alpha, beta, gma, dlt, out);
}